// downWalkBlock_65687229826153
// MI455X (gfx1250) — compile-verified
//
#include <hip/hip_runtime.h>
#include <stdint.h>

// Problem constants (match reference)
#define BB   8
#define NPT  2048
#define SPT  512
#define C_IN 128
#define C_OUT 256
#define BNSZ 64
#define KNN  8
#define EPSV 1e-5f
#define RAD2 0.04f   // 0.2^2

typedef __attribute__((ext_vector_type(2))) float v2f;
typedef __attribute__((ext_vector_type(8))) float v8f;

// ---------------------------------------------------------------------------
// WMMA fp32 GEMM: one wave computes a 16(M) x 64(N) strip of
// C = A(MxK,row-major) * B(KxN) using V_WMMA_F32_16X16X4_F32.
// The A fragment (16x4) is loaded once per K-step and reused across the four
// 4x16 B tiles -> 4 back-to-back wmma per K-step, A traffic / 4.
// A lane layout (16x4 f32): lane L holds row M=L&15; VGPR0 = K = kk+2*(L>>4),
// VGPR1 = K+1.  B (4x16): lane L holds col N=L&15, same K split.  C/D: lane L
// col N=L&15, VGPR v row M = v + 8*(L>>4).
// BT==0: B element = Bm[k*ldb + n]   (row-major KxN, e.g. feature matrix)
// BT==1: B element = Bm[n*ldb + k]   (weight matrix W[n][k], i.e. B = W^T)
// ---------------------------------------------------------------------------
template <int BT>
__global__ __launch_bounds__(32)
void wmma_gemm_f32(const float* __restrict__ A, int lda, long sA,
                   const float* __restrict__ Bm, int ldb, long sB,
                   float* __restrict__ C, int ldc, long sC, int Kdim) {
  const int lane = threadIdx.x;
  const int hi   = lane >> 4;
  const int l15  = lane & 15;
  const int m0   = blockIdx.y * 16;
  const int n0   = blockIdx.x * 64;
  A  += (long)blockIdx.z * sA;
  Bm += (long)blockIdx.z * sB;
  C  += (long)blockIdx.z * sC;

  v8f acc[4];
#pragma unroll
  for (int t = 0; t < 4; t++) acc[t] = (v8f){0.f,0.f,0.f,0.f,0.f,0.f,0.f,0.f};

  const float* ap = A + (long)(m0 + l15) * lda;
  for (int kk = 0; kk < Kdim; kk += 4) {
    const int kb = kk + 2*hi;
    v2f a;
    a.x = ap[kb];
    a.y = ap[kb + 1];
#pragma unroll
    for (int t = 0; t < 4; t++) {
      const int nc = n0 + 16*t + l15;
      v2f b;
      if (BT == 0) {
        b.x = Bm[(long)kb * ldb + nc];
        b.y = Bm[(long)(kb + 1) * ldb + nc];
      } else {
        const float* bp = Bm + (long)nc * ldb;
        b.x = bp[kb];
        b.y = bp[kb + 1];
      }
      acc[t] = __builtin_amdgcn_wmma_f32_16x16x4_f32(
          /*neg_a=*/false, a, /*neg_b=*/false, b,
          /*c_mod=*/(short)0, acc[t], /*reuse_a=*/false, /*reuse_b=*/false);
    }
  }
#pragma unroll
  for (int t = 0; t < 4; t++) {
#pragma unroll
    for (int v = 0; v < 8; v++) {
      const int m = m0 + v + 8*hi;
      C[(long)m * ldc + n0 + 16*t + l15] = acc[t][v];
    }
  }
}

// ---------------------------------------------------------------------------
// BatchNorm statistics: one block per channel.  Element index of sample i of
// channel c is  c*cs + (i/inner)*os + (i%inner)*is.
// Stores scale = g*rsqrt(var+eps) at ss[c], shift = beta - mean*scale at ss[C+c].
// ---------------------------------------------------------------------------
__global__ __launch_bounds__(256)
void bn_stats(const float* __restrict__ x, long cs, long inner, long os, long is,
              int count, const float* __restrict__ g, const float* __restrict__ bta,
              float* __restrict__ ss, int C) {
  __shared__ float s1[256], s2[256];
  const int c = blockIdx.x;
  const int tid = threadIdx.x;
  float sum = 0.f, sq = 0.f;
  for (int i = tid; i < count; i += 256) {
    const long idx = (long)c * cs + (long)(i / inner) * os + (long)(i % inner) * is;
    const float v = x[idx];
    sum += v; sq += v * v;
  }
  s1[tid] = sum; s2[tid] = sq;
  __syncthreads();
  for (int st = 128; st > 0; st >>= 1) {
    if (tid < st) { s1[tid] += s1[tid + st]; s2[tid] += s2[tid + st]; }
    __syncthreads();
  }
  if (tid == 0) {
    const float m   = s1[0] / (float)count;
    const float var = s2[0] / (float)count - m * m;
    const float sc  = g[c] * rsqrtf(var + EPSV);
    ss[c]     = sc;
    ss[C + c] = bta[c] - m * sc;
  }
}

// bn1 apply + relu, in place on x1 laid out (b, c=64, n=2048)
__global__ void bn1_apply(float* x1, const float* __restrict__ ss) {
  const long gid = (long)blockIdx.x * blockDim.x + threadIdx.x;
  const int c = (int)((gid >> 11) & 63);
  x1[gid] = fmaxf(x1[gid] * ss[c] + ss[64 + c], 0.f);
}

// column-layout BN apply + relu, in place (rows x 64)
__global__ void wbn_apply(float* a, const float* __restrict__ ss) {
  const long gid = (long)blockIdx.x * blockDim.x + threadIdx.x;
  const int c = (int)(gid & 63);
  a[gid] = fmaxf(a[gid] * ss[c] + ss[64 + c], 0.f);
}

// bn2 apply + relu + multiply by feat; y2 is (row=b*2048+n, c=128); write x2 (b,c,n)
__global__ void bn2_mulfeat(const float* __restrict__ y2, const float* __restrict__ feat,
                            const float* __restrict__ ss, float* __restrict__ x2) {
  const long gid = (long)blockIdx.x * blockDim.x + threadIdx.x;
  const long row = gid >> 7;
  const int  c   = (int)(gid & 127);
  const long b   = row >> 11;
  const long n   = row & 2047;
  const long f   = (b * C_IN + c) * (long)NPT + n;
  x2[f] = fmaxf(y2[gid] * ss[c] + ss[128 + c], 0.f) * feat[f];
}

// knn on first two coordinates of xyz; 8 smallest distances, lowest-index ties
__global__ void knn_kernel(const float* __restrict__ xyz, int* __restrict__ idx) {
  const int gid = blockIdx.x * blockDim.x + threadIdx.x;
  if (gid >= BB * NPT) return;
  const int b = gid / NPT, n = gid % NPT;
  const float* px = xyz + (long)b * 3 * NPT;
  const float* py = px + NPT;
  const float xn = px[n], yn = py[n];
  const float sqn = xn * xn + yn * yn;
  float bd[KNN]; int bi[KNN];
#pragma unroll
  for (int k = 0; k < KNN; k++) { bd[k] = 3.0e38f; bi[k] = 0; }
  for (int m = 0; m < NPT; m++) {
    const float xm = px[m], ym = py[m];
    const float d = sqn + (xm * xm + ym * ym) - 2.f * (xn * xm + yn * ym);
    if (d < bd[KNN - 1]) {
      int p = KNN - 1;
      while (p > 0 && d < bd[p - 1]) { bd[p] = bd[p - 1]; bi[p] = bi[p - 1]; p--; }
      bd[p] = d; bi[p] = m;
    }
  }
#pragma unroll
  for (int k = 0; k < KNN; k++) idx[(long)gid * KNN + k] = bi[k];
}

// lap + affine contraction, reproducing the reference reshape permutation:
// m = (c%8)*16384 + n*8 + k ; sum over k' of affine_w[k'] * x1[b, 8k'+(c%8), idx[b,n,k]]
__global__ void lap_aff_kernel(const float* __restrict__ x1, const int* __restrict__ idx,
                               const float* __restrict__ aw, const float* __restrict__ ab,
                               float* __restrict__ aff) {
  const long gid = (long)blockIdx.x * blockDim.x + threadIdx.x;
  const int b = (int)(gid >> 17);          // / 131072
  const int m = (int)(gid & 131071);
  const int r = m >> 14;                   // c % 8
  const int j = m & 16383;
  const int n = j >> 3;
  const int k = j & 7;
  const int pt = idx[((long)b * NPT + n) * KNN + k];
  const float* xb = x1 + (long)b * BNSZ * NPT;
  float s = 0.f;
#pragma unroll
  for (int kp = 0; kp < 8; kp++) s += aw[kp] * xb[(8 * kp + r) * NPT + pt];
  aff[gid] = s + ab[0];
}

// Furthest point sampling: one block per batch, 256 threads, 8 points/thread,
// distances in registers, LDS argmax (first-max-index tie rule = jnp.argmax).
__global__ __launch_bounds__(256)
void fps_kernel(const float* __restrict__ xyz, float* __restrict__ newxyz) {
  __shared__ float sv[256];
  __shared__ int   si[256];
  __shared__ float c3[3];
  const int b = blockIdx.x, tid = threadIdx.x;
  const float* xb = xyz + (long)b * 3 * NPT;
  float px[8], py[8], pz[8], dist[8];
#pragma unroll
  for (int i = 0; i < 8; i++) {
    const int n = tid * 8 + i;
    px[i] = xb[n]; py[i] = xb[NPT + n]; pz[i] = xb[2 * NPT + n];
    dist[i] = 1e10f;
  }
  int far = 0;
  for (int it = 0; it < SPT; it++) {
    if (tid == 0) {
      const float cx = xb[far], cy = xb[NPT + far], cz = xb[2 * NPT + far];
      float* o = newxyz + ((long)b * SPT + it) * 3;
      o[0] = cx; o[1] = cy; o[2] = cz;
      c3[0] = cx; c3[1] = cy; c3[2] = cz;
    }
    __syncthreads();
    const float cx = c3[0], cy = c3[1], cz = c3[2];
    float bestv = -1.f; int besti = 0;
#pragma unroll
    for (int i = 0; i < 8; i++) {
      const float dx = px[i] - cx, dy = py[i] - cy, dz = pz[i] - cz;
      const float d = dx * dx + dy * dy + dz * dz;
      if (d < dist[i]) dist[i] = d;
      if (dist[i] > bestv) { bestv = dist[i]; besti = tid * 8 + i; }
    }
    sv[tid] = bestv; si[tid] = besti;
    __syncthreads();
    for (int st = 128; st > 0; st >>= 1) {
      if (tid < st) {
        if (sv[tid + st] > sv[tid] ||
            (sv[tid + st] == sv[tid] && si[tid + st] < si[tid])) {
          sv[tid] = sv[tid + st]; si[tid] = si[tid + st];
        }
      }
      __syncthreads();
    }
    far = si[0];
    __syncthreads();
  }
}

// Ball query: first K indices (ascending) with d <= r^2, padded with the first.
__global__ void ballq_kernel(const float* __restrict__ xyz,
                             const float* __restrict__ newxyz, int* __restrict__ gi) {
  const int gid = blockIdx.x * blockDim.x + threadIdx.x;
  if (gid >= BB * SPT) return;
  const int b = gid / SPT;
  const float* xb = xyz + (long)b * 3 * NPT;
  const float nx = newxyz[(long)gid * 3 + 0];
  const float ny = newxyz[(long)gid * 3 + 1];
  const float nz = newxyz[(long)gid * 3 + 2];
  const float sqn = nx * nx + ny * ny + nz * nz;
  int out[KNN]; int cnt = 0;
  for (int n = 0; n < NPT && cnt < KNN; n++) {
    const float x = xb[n], y = xb[NPT + n], z = xb[2 * NPT + n];
    const float d = sqn + (x * x + y * y + z * z) - 2.f * (nx * x + ny * y + nz * z);
    if (!(d > RAD2)) out[cnt++] = n;
  }
  const int first = (cnt > 0) ? out[0] : (NPT - 1);  // JAX OOB gather clips
  for (int k = cnt; k < KNN; k++) out[k] = first;
#pragma unroll
  for (int k = 0; k < KNN; k++) gi[(long)gid * KNN + k] = out[k];
}

// Gather grouped features as GEMM B-matrix: gf[c, col], col = (b*S+s)*K + k
__global__ void gather_kernel(const float* __restrict__ x2, const int* __restrict__ gi,
                              float* __restrict__ gf) {
  const long gid = (long)blockIdx.x * blockDim.x + threadIdx.x;
  const int c   = (int)(gid >> 15);
  const int col = (int)(gid & 32767);
  const int bs  = col >> 3;
  const int b   = bs >> 9;
  const int pt  = gi[col];
  gf[gid] = x2[((long)b * C_IN + c) * NPT + pt];
}

// cbn apply + relu + max over K; h1 is (o=256, col=32768); hmax is (bs, o)
__global__ void cbn_max(const float* __restrict__ h1, const float* __restrict__ ss,
                        float* __restrict__ hmax) {
  const long gid = (long)blockIdx.x * blockDim.x + threadIdx.x;
  const int bs = (int)(gid >> 8);
  const int o  = (int)(gid & 255);
  const float sc = ss[o], sh = ss[256 + o];
  const float* hp = h1 + (long)o * 32768 + (long)bs * KNN;
  float m = 0.f;  // relu output >= 0
#pragma unroll
  for (int k = 0; k < KNN; k++) m = fmaxf(m, fmaxf(hp[k] * sc + sh, 0.f));
  hmax[gid] = m;
}

// bn3 apply + relu, store to output as (b, o, s)
__global__ void bn3_store(const float* __restrict__ o3, const float* __restrict__ ss,
                          float* __restrict__ oh) {
  const long gid = (long)blockIdx.x * blockDim.x + threadIdx.x;
  const int row = (int)(gid >> 8);   // b*512 + s
  const int o   = (int)(gid & 255);
  const int b   = row >> 9;
  const int s   = row & 511;
  const float v = fmaxf(o3[gid] * ss[o] + ss[256 + o], 0.f);
  oh[((long)b * C_OUT + o) * SPT + s] = v;
}

// ---------------------------------------------------------------------------
extern "C" void kernel_launch(void* const* d_in, const int* in_sizes, int n_in,
                              void* d_out, int out_size, void* d_ws, size_t ws_size,
                              hipStream_t stream) {
  const float* xyz         = (const float*)d_in[0];
  const float* feat        = (const float*)d_in[1];
  const float* fc1_w       = (const float*)d_in[2];
  const float* bn1_g       = (const float*)d_in[4];
  const float* bn1_b       = (const float*)d_in[5];
  const float* affine_w    = (const float*)d_in[6];
  const float* affine_b    = (const float*)d_in[7];
  const float* transform_w = (const float*)d_in[8];
  const float* wbn_g       = (const float*)d_in[10];
  const float* wbn_b       = (const float*)d_in[11];
  const float* fc2_w       = (const float*)d_in[12];
  const float* bn2_g       = (const float*)d_in[14];
  const float* bn2_b       = (const float*)d_in[15];
  const float* conv_w      = (const float*)d_in[16];
  const float* cbn_g       = (const float*)d_in[18];
  const float* cbn_b       = (const float*)d_in[19];
  const float* fc3_w       = (const float*)d_in[20];
  const float* bn3_g       = (const float*)d_in[22];
  const float* bn3_b       = (const float*)d_in[23];
  float* out = (float*)d_out;
  // All linear biases (fc1_b, transform_b, fc2_b, conv_b, fc3_b) feed directly
  // into a BatchNorm: the per-channel constant is removed by mean subtraction
  // (and they are zeros in setup_inputs), so they are omitted.

  // Workspace layout (floats)
  float* W    = (float*)d_ws;
  float* x1   = W;                 // (B,64,N)    1,048,576
  float* aff  = x1   + 1048576;    // (B*N,64)    1,048,576
  float* aff2 = aff  + 1048576;    // (B*N,64)    1,048,576
  float* y2   = aff2 + 1048576;    // (B*N,128)   2,097,152
  float* x2   = y2   + 2097152;    // (B,128,N)   2,097,152
  float* gf   = x2   + 2097152;    // (128,32768) 4,194,304
  float* h1   = gf   + 4194304;    // (256,32768) 8,388,608
  float* hmax = h1   + 8388608;    // (B*S,256)   1,048,576
  float* o3   = hmax + 1048576;    // (B*S,256)   1,048,576
  float* ss   = o3   + 1048576;    // scale/shift 512
  int* knn_idx = (int*)(ss + 512); // (B,N,K)     131,072
  int* gi      = knn_idx + 131072; // (B,S,K)     32,768

  // 1) fc1: x1[b,o,n] = fc1_w(64x128) @ feat[b](128x2048)
  wmma_gemm_f32<0><<<dim3(NPT/64, BNSZ/16, BB), 32, 0, stream>>>(
      fc1_w, C_IN, 0, feat, NPT, (long)C_IN*NPT, x1, NPT, (long)BNSZ*NPT, C_IN);
  bn_stats<<<BNSZ, 256, 0, stream>>>(x1, NPT, NPT, (long)BNSZ*NPT, 1, BB*NPT,
                                     bn1_g, bn1_b, ss, BNSZ);
  bn1_apply<<<1048576/256, 256, 0, stream>>>(x1, ss);

  // 2) knn on xyz[:,0:2,:]
  knn_kernel<<<(BB*NPT)/64, 64, 0, stream>>>(xyz, knn_idx);

  // 3) lap gather + affine contraction -> aff (B*N, 64)
  lap_aff_kernel<<<1048576/256, 256, 0, stream>>>(x1, knn_idx, affine_w, affine_b, aff);

  // 4) transform: aff2 = aff @ transform_w^T   (16384 x 64 x 64)
  wmma_gemm_f32<1><<<dim3(BNSZ/64, (BB*NPT)/16, 1), 32, 0, stream>>>(
      aff, BNSZ, 0, transform_w, BNSZ, 0, aff2, BNSZ, 0, BNSZ);
  bn_stats<<<BNSZ, 256, 0, stream>>>(aff2, 1, BB*NPT, 0, BNSZ, BB*NPT,
                                     wbn_g, wbn_b, ss, BNSZ);
  wbn_apply<<<1048576/256, 256, 0, stream>>>(aff2, ss);

  // 5) fc2: y2 = aff2 @ fc2_w^T (16384 x 64 x 128), then bn2 + relu + *feat -> x2
  wmma_gemm_f32<1><<<dim3(C_IN/64, (BB*NPT)/16, 1), 32, 0, stream>>>(
      aff2, BNSZ, 0, fc2_w, BNSZ, 0, y2, C_IN, 0, BNSZ);
  bn_stats<<<C_IN, 256, 0, stream>>>(y2, 1, BB*NPT, 0, C_IN, BB*NPT,
                                     bn2_g, bn2_b, ss, C_IN);
  bn2_mulfeat<<<2097152/256, 256, 0, stream>>>(y2, feat, ss, x2);

  // 6) FPS (writes newxyz straight to d_out), then ball query
  fps_kernel<<<BB, 256, 0, stream>>>(xyz, out);
  ballq_kernel<<<(BB*SPT)/64, 64, 0, stream>>>(xyz, out, gi);

  // 7) gather grouped features -> gf (128 x 32768)
  gather_kernel<<<4194304/256, 256, 0, stream>>>(x2, gi, gf);

  // 8) conv: h1 = conv_w(256x128) @ gf(128x32768) — the dominant GEMM (2.1 GFLOP)
  wmma_gemm_f32<0><<<dim3(32768/64, C_OUT/16, 1), 32, 0, stream>>>(
      conv_w, C_IN, 0, gf, 32768, 0, h1, 32768, 0, C_IN);
  bn_stats<<<C_OUT, 256, 0, stream>>>(h1, 32768, 32768, 0, 1, 32768,
                                      cbn_g, cbn_b, ss, C_OUT);
  cbn_max<<<1048576/256, 256, 0, stream>>>(h1, ss, hmax);

  // 9) fc3: o3 = hmax(4096x256) @ fc3_w^T, bn3 + relu -> output (B, COUT, S)
  wmma_gemm_f32<1><<<dim3(C_OUT/64, (BB*SPT)/16, 1), 32, 0, stream>>>(
      hmax, C_OUT, 0, fc3_w, C_OUT, 0, o3, C_OUT, 0, C_OUT);
  bn_stats<<<C_OUT, 256, 0, stream>>>(o3, 1, BB*SPT, 0, C_OUT, BB*SPT,
                                      bn3_g, bn3_b, ss, C_OUT);
  bn3_store<<<1048576/256, 256, 0, stream>>>(o3, ss, out + (long)BB*SPT*3);
}